// Eeg_csp_38869454029730
// MI455X (gfx1250) — compile-verified
//
#include <hip/hip_runtime.h>
#include <math.h>

typedef __attribute__((ext_vector_type(2))) float v2f;
typedef __attribute__((ext_vector_type(8))) float v8f;
typedef __attribute__((ext_vector_type(4))) unsigned int u32x4;
typedef __attribute__((ext_vector_type(4))) int i32x4;
typedef __attribute__((ext_vector_type(8))) int i32x8;

#define Bc 8
#define Sc 1024
#define Ec 256
#define Hc 8
#define Dc 32
#define LKc 512     // half-sequence (keys per attention call)
#define LCc 2048    // concat length

// -------- workspace layout (float offsets) --------
#define WS_Q    ((size_t)0)
#define WS_K    ((size_t)2097152)
#define WS_V    ((size_t)4194304)
#define WS_O    ((size_t)6291456)     // 4*8*512*256
#define WS_LN   ((size_t)10485760)    // 8*2048*256
#define WS_CONV ((size_t)14680064)    // 8*2048*256
#define WS_BNM  ((size_t)18874368)    // 256
#define WS_BNV  ((size_t)18874624)    // 256

// -------- output layout (float offsets) --------
#define OUT_XSRC   ((size_t)0)          // 8*1024*256
#define OUT_LOGITS ((size_t)2097152)    // 32
#define OUT_ATTN   ((size_t)2097184)    // a11,a22,a12,a21 each 8*8*512*512
#define ATTN_SZ    ((size_t)16777216)

// =====================================================================
// Tensor Data Mover helpers (D# built per CDNA5 ISA)
// =====================================================================
__device__ __forceinline__ u32x4 tdm_g0(unsigned lds_byte_addr,
                                        unsigned long long gaddr) {
    u32x4 g;
    g.x = 1u;                                   // count=1, user descriptor
    g.y = lds_byte_addr;                        // lds_addr[31:0]
    g.z = (unsigned)(gaddr & 0xFFFFFFFFu);      // global_addr[31:0]
    g.w = (unsigned)((gaddr >> 32) & 0x01FFFFFFu) | (2u << 30); // addr[56:32] | type=2
    return g;
}

// 2-D tile (or 1-D when tile_d1 == 0); 4-byte elements; strides in elements
__device__ __forceinline__ i32x8 tdm_g1(unsigned tensor_d0, unsigned tensor_d1,
                                        unsigned tile_d0, unsigned tile_d1,
                                        unsigned long long stride0) {
    i32x8 g = {};
    g[0] = (int)(2u << 16);                                       // data_size = 4B
    g[1] = (int)((tensor_d0 & 0xFFFFu) << 16);                    // tensor_dim0 lo
    g[2] = (int)(((tensor_d0 >> 16) & 0xFFFFu) |
                 ((tensor_d1 & 0xFFFFu) << 16));                  // d0 hi | d1 lo
    g[3] = (int)(((tensor_d1 >> 16) & 0xFFFFu) |
                 ((tile_d0 & 0xFFFFu) << 16));                    // d1 hi | tile_d0
    g[4] = (int)(tile_d1 & 0xFFFFu);                              // tile_d1 | tile_d2=0
    g[5] = (int)(stride0 & 0xFFFFFFFFu);                          // dim0_stride lo
    g[6] = (int)((stride0 >> 32) & 0xFFFFu);                      // dim0_stride hi
    g[7] = 0;
    return g;
}

__device__ __forceinline__ void tdm_load(u32x4 g0, i32x8 g1) {
    i32x4 z4 = {};
#if __clang_major__ >= 23
    i32x8 z8 = {};
    __builtin_amdgcn_tensor_load_to_lds(g0, g1, z4, z4, z8, 0);
#else
    __builtin_amdgcn_tensor_load_to_lds(g0, g1, z4, z4, 0);
#endif
}

__device__ __forceinline__ void tdm_store(u32x4 g0, i32x8 g1) {
    i32x4 z4 = {};
#if __clang_major__ >= 23
    i32x8 z8 = {};
    __builtin_amdgcn_tensor_store_from_lds(g0, g1, z4, z4, z8, 0);
#else
    __builtin_amdgcn_tensor_store_from_lds(g0, g1, z4, z4, 0);
#endif
}

// =====================================================================
// K1: Q/K/V projection  C[m,n] = sum_k X[m,k]*W[n,k]   (one wave / tile)
// =====================================================================
__global__ __launch_bounds__(32) void qkv_kernel(
    const float* __restrict__ x, const float* __restrict__ Wq,
    const float* __restrict__ Wk, const float* __restrict__ Wv,
    float* __restrict__ ws)
{
    const int lane  = threadIdx.x;
    const int row16 = lane & 15;
    const int khalf = lane >> 4;
    const int n0 = blockIdx.x * 16;
    const int m0 = blockIdx.y * 16;
    const int which = blockIdx.z;
    const float* W = (which == 0) ? Wq : (which == 1) ? Wk : Wv;
    float* dst = ws + (size_t)which * (Bc * Sc * Ec);

    const float* ap = x + (size_t)(m0 + row16) * Ec + 2 * khalf;
    const float* bp = W + (size_t)(n0 + row16) * Ec + 2 * khalf;
    v8f acc = {};
    for (int k = 0; k < Ec; k += 4) {
        v2f a; a.x = ap[k]; a.y = ap[k + 1];
        v2f b; b.x = bp[k]; b.y = bp[k + 1];
        acc = __builtin_amdgcn_wmma_f32_16x16x4_f32(false, a, false, b,
                                                    (short)0, acc, false, false);
    }
    const int crow = m0 + khalf * 8;
    const int ccol = n0 + row16;
#pragma unroll
    for (int i = 0; i < 8; ++i)
        dst[(size_t)(crow + i) * Ec + ccol] = acc[i];
}

// =====================================================================
// K2: attention per (call, b*H+h, q-tile of 16).
//     TDM prefetches the 512x32 V head-tile into LDS while scores +
//     softmax run; probabilities leave through TENSOR_STORE_FROM_LDS.
// =====================================================================
__global__ __launch_bounds__(256) void attn_kernel(
    float* __restrict__ ws, float* __restrict__ out)
{
    __shared__ float sc[16][512];     // score / probability tile (32 KB)
    __shared__ float vtile[512][32];  // V head tile, TDM destination (64 KB)
    __shared__ float oacc[16][32];    // P@V accumulator
    __shared__ float red[16][17];
    __shared__ float rowmax[16];
    __shared__ float rowsum[16];

    const int c  = blockIdx.z;                 // 0:o12 1:o21 2:o22 3:o11
    const int bh = blockIdx.y;
    const int b  = bh >> 3, h = bh & 7;
    const int qt = blockIdx.x;
    const int qHalf  = (c == 0 || c == 2) ? 1 : 0;
    const int kvHalf = (c == 1 || c == 2) ? 1 : 0;
    const int slot   = (c == 0) ? 2 : (c == 1) ? 3 : (c == 2) ? 1 : 0; // a11,a22,a12,a21

    const float* Q = ws + WS_Q;
    const float* K = ws + WS_K;
    const float* V = ws + WS_V;

    const int tid  = threadIdx.x;
    const int wave = tid >> 5, lane = tid & 31;
    const int row16 = lane & 15, khalf = lane >> 4;

    const size_t qbase  = ((size_t)b * Sc + (size_t)qHalf * LKc + qt * 16) * Ec + h * Dc;
    const size_t kvbase = ((size_t)b * Sc + (size_t)kvHalf * LKc) * Ec + h * Dc;

    // ---- kick off async V-tile DMA (512 rows x 32 floats, row stride 256) ----
    if (wave == 0) {
        u32x4 g0 = tdm_g0((unsigned)(uintptr_t)&vtile[0][0],
                          (unsigned long long)(uintptr_t)(V + kvbase));
        i32x8 g1 = tdm_g1(/*tensor_d0=*/Ec, /*tensor_d1=*/LKc,
                          /*tile_d0=*/Dc, /*tile_d1=*/LKc, /*stride0=*/Ec);
        tdm_load(g0, g1);
    }

    // ---- scores: each wave does 4 of the 32 key-tiles ----
    for (int nt = wave * 4; nt < wave * 4 + 4; ++nt) {
        const float* ap = Q + qbase + (size_t)row16 * Ec + 2 * khalf;
        const float* bp = K + kvbase + (size_t)(nt * 16 + row16) * Ec + 2 * khalf;
        v8f acc = {};
#pragma unroll
        for (int kk = 0; kk < Dc; kk += 4) {
            v2f a;  a.x  = ap[kk]; a.y  = ap[kk + 1];
            v2f bb; bb.x = bp[kk]; bb.y = bp[kk + 1];
            acc = __builtin_amdgcn_wmma_f32_16x16x4_f32(false, a, false, bb,
                                                        (short)0, acc, false, false);
        }
#pragma unroll
        for (int i = 0; i < 8; ++i)
            sc[khalf * 8 + i][nt * 16 + row16] = acc[i];
    }
    __syncthreads();

    // ---- softmax: 16 threads per row, 32 cols each ----
    const int r = tid >> 4, seg = tid & 15;
    float lmax = -3.4e38f;
    for (int j = 0; j < 32; ++j) lmax = fmaxf(lmax, sc[r][seg * 32 + j]);
    red[r][seg] = lmax;
    __syncthreads();
    if (seg == 0) {
        float m = red[r][0];
        for (int j = 1; j < 16; ++j) m = fmaxf(m, red[r][j]);
        rowmax[r] = m;
    }
    __syncthreads();
    float m = rowmax[r], lsum = 0.f;
    for (int j = 0; j < 32; ++j) {
        float e = __expf(sc[r][seg * 32 + j] - m);
        sc[r][seg * 32 + j] = e;
        lsum += e;
    }
    red[r][seg] = lsum;
    __syncthreads();
    if (seg == 0) {
        float s = 0.f;
        for (int j = 0; j < 16; ++j) s += red[r][j];
        rowsum[r] = s;
    }
    __syncthreads();
    {
        float inv = 1.0f / rowsum[r];
        for (int j = 0; j < 32; ++j)
            sc[r][seg * 32 + j] *= inv;
    }
    ((float*)oacc)[tid] = 0.f;
    ((float*)oacc)[tid + 256] = 0.f;
    __syncthreads();   // probs final in LDS, oacc zeroed, all waves synced

    // ---- stream the 16x512 probability tile to d_out via TDM (contiguous) ----
    if (wave == 0) {
        float* aout = out + OUT_ATTN + slot * ATTN_SZ
                    + (((size_t)(b * Hc + h) * LKc) + qt * 16) * LKc;
        u32x4 g0 = tdm_g0((unsigned)(uintptr_t)&sc[0][0],
                          (unsigned long long)(uintptr_t)aout);
        i32x8 g1 = tdm_g1(/*tensor_d0=*/16 * LKc, /*tensor_d1=*/0,
                          /*tile_d0=*/16 * LKc, /*tile_d1=*/0,
                          /*stride0=*/16 * LKc);
        tdm_store(g0, g1);
    }
    // wave 0: oldest tensor op (V load) must be done; others pass trivially
    __builtin_amdgcn_s_wait_tensorcnt((short)1);
    __syncthreads();

    // ---- P @ V from LDS: waves 0-3 -> n-tile 0, 4-7 -> n-tile 1; K split x4 ----
    {
        const int nt2 = wave >> 2;
        const int ks  = wave & 3;
        v8f acc = {};
        for (int kk = ks * 128; kk < ks * 128 + 128; kk += 4) {
            v2f a;  a.x = sc[row16][kk + 2 * khalf];
                    a.y = sc[row16][kk + 2 * khalf + 1];
            v2f bb; bb.x = vtile[kk + 2 * khalf]    [nt2 * 16 + row16];
                    bb.y = vtile[kk + 2 * khalf + 1][nt2 * 16 + row16];
            acc = __builtin_amdgcn_wmma_f32_16x16x4_f32(false, a, false, bb,
                                                        (short)0, acc, false, false);
        }
#pragma unroll
        for (int i = 0; i < 8; ++i)
            atomicAdd(&oacc[khalf * 8 + i][nt2 * 16 + row16], acc[i]);
    }
    __syncthreads();

    // ---- write O tile to scratch: ((c*B+b)*512 + row)*E + h*32 + d ----
#pragma unroll
    for (int rep = 0; rep < 2; ++rep) {
        int e = tid + rep * 256;
        int rr = e >> 5, d = e & 31;
        ws[WS_O + (((size_t)(c * Bc + b) * LKc) + qt * 16 + rr) * Ec + h * Dc + d] =
            oacc[rr][d];
    }
    // drain the probability TDM store before the wave retires
    __builtin_amdgcn_s_wait_tensorcnt((short)0);
}

// =====================================================================
// K3: LayerNorm over E=256 (one block per row of the 2S concat)
// =====================================================================
__global__ __launch_bounds__(256) void ln_kernel(
    float* __restrict__ ws, const float* __restrict__ g, const float* __restrict__ bt)
{
    __shared__ float red[256];
    const int gr = blockIdx.x;          // 0..16383
    const int b = gr >> 11, lr = gr & 2047;
    const int c = lr >> 9,  r  = lr & 511;
    const int e = threadIdx.x;
    float v = ws[WS_O + (((size_t)(c * Bc + b) * LKc) + r) * Ec + e];

    red[e] = v; __syncthreads();
    for (int s = 128; s > 0; s >>= 1) { if (e < s) red[e] += red[e + s]; __syncthreads(); }
    float mu = red[0] * (1.0f / 256.0f);
    __syncthreads();
    float d = v - mu;
    red[e] = d * d; __syncthreads();
    for (int s = 128; s > 0; s >>= 1) { if (e < s) red[e] += red[e + s]; __syncthreads(); }
    float var = red[0] * (1.0f / 256.0f);
    ws[WS_LN + ((size_t)b * LCc + lr) * Ec + e] =
        d * rsqrtf(var + 1e-5f) * g[e] + bt[e];
}

// =====================================================================
// K4: Conv1d(k=3,p=1) as WMMA GEMM over K = 3*256 (one wave / tile)
// =====================================================================
__global__ __launch_bounds__(32) void conv_kernel(
    const float* __restrict__ cw, const float* __restrict__ cb,
    float* __restrict__ ws)
{
    const int lane  = threadIdx.x;
    const int row16 = lane & 15, khalf = lane >> 4;
    const int n0 = blockIdx.x * 16;     // out channel tile
    const int t0 = blockIdx.y * 16;     // time tile
    const int b  = blockIdx.z;
    const float* ln = ws + WS_LN + (size_t)b * LCc * Ec;

    v8f acc = {};
    for (int dt = 0; dt < 3; ++dt) {
        const int t = t0 + row16 + dt - 1;
        const bool ok = (t >= 0) && (t < LCc);
        const float* ap = ln + (size_t)(ok ? t : 0) * Ec + 2 * khalf;
        const float* bp = cw + ((size_t)(n0 + row16) * Ec + 2 * khalf) * 3 + dt;
        for (int k = 0; k < Ec; k += 4) {
            v2f a;
            a.x = ok ? ap[k]     : 0.f;
            a.y = ok ? ap[k + 1] : 0.f;
            v2f bb; bb.x = bp[(size_t)k * 3]; bb.y = bp[(size_t)k * 3 + 3];
            acc = __builtin_amdgcn_wmma_f32_16x16x4_f32(false, a, false, bb,
                                                        (short)0, acc, false, false);
        }
    }
    const int crow = khalf * 8;
    const int ccol = n0 + row16;
    const float bias = cb[ccol];
#pragma unroll
    for (int i = 0; i < 8; ++i)
        ws[WS_CONV + ((size_t)b * LCc + t0 + crow + i) * Ec + ccol] = acc[i] + bias;
}

// =====================================================================
// K5: BatchNorm batch statistics per channel (train mode)
// =====================================================================
__global__ __launch_bounds__(256) void bnstat_kernel(float* __restrict__ ws)
{
    __shared__ float rs[256], rq[256];
    const int co = blockIdx.x, tid = threadIdx.x;
    float s = 0.f, q = 0.f;
    for (int i = tid; i < Bc * LCc; i += 256) {
        float v = ws[WS_CONV + (size_t)i * Ec + co];
        s += v; q += v * v;
    }
    rs[tid] = s; rq[tid] = q; __syncthreads();
    for (int st = 128; st > 0; st >>= 1) {
        if (tid < st) { rs[tid] += rs[tid + st]; rq[tid] += rq[tid + st]; }
        __syncthreads();
    }
    if (tid == 0) {
        float mean = rs[0] / (float)(Bc * LCc);
        ws[WS_BNM + co] = mean;
        ws[WS_BNV + co] = rq[0] / (float)(Bc * LCc) - mean * mean;
    }
}

// =====================================================================
// K6: BN normalize + exact GELU + MaxPool(k3,s2,p1) -> x_src in d_out
// =====================================================================
__global__ __launch_bounds__(256) void pool_kernel(
    const float* __restrict__ bng, const float* __restrict__ bnb,
    const float* __restrict__ ws, float* __restrict__ out)
{
    const int co = threadIdx.x;
    const int j  = blockIdx.x;   // 0..1023 pooled position
    const int b  = blockIdx.y;
    float scale = bng[co] * rsqrtf(ws[WS_BNV + co] + 1e-5f);
    float shift = bnb[co] - ws[WS_BNM + co] * scale;
    float best = -3.4e38f;
#pragma unroll
    for (int dt = -1; dt <= 1; ++dt) {
        int t = 2 * j + dt;
        if (t < 0 || t >= LCc) continue;
        float v = ws[WS_CONV + ((size_t)b * LCc + t) * Ec + co] * scale + shift;
        float g = 0.5f * v * (1.0f + erff(v * 0.70710678118654752f));
        best = fmaxf(best, g);
    }
    out[OUT_XSRC + ((size_t)b * Sc + j) * Ec + co] = best;
}

// =====================================================================
// K7: GAP over length + linear head -> logits
// =====================================================================
__global__ __launch_bounds__(256) void head_kernel(
    const float* __restrict__ ow, const float* __restrict__ ob,
    float* __restrict__ out)
{
    __shared__ float gap[256];
    const int co = threadIdx.x, b = blockIdx.x;
    float s = 0.f;
    for (int j = 0; j < Sc; ++j)
        s += out[OUT_XSRC + ((size_t)b * Sc + j) * Ec + co];
    gap[co] = s * (1.0f / (float)Sc);
    __syncthreads();
    if (co < 4) {
        float acc = ob[co];
        for (int k = 0; k < Ec; ++k) acc += gap[k] * ow[co * Ec + k];
        out[OUT_LOGITS + b * 4 + co] = acc;
    }
}

extern "C" void kernel_launch(void* const* d_in, const int* in_sizes, int n_in,
                              void* d_out, int out_size, void* d_ws, size_t ws_size,
                              hipStream_t stream) {
    (void)in_sizes; (void)n_in; (void)out_size; (void)ws_size;
    const float* x      = (const float*)d_in[0];
    const float* Wk     = (const float*)d_in[1];
    const float* Wq     = (const float*)d_in[2];
    const float* Wv     = (const float*)d_in[3];
    const float* ln_g   = (const float*)d_in[4];
    const float* ln_b   = (const float*)d_in[5];
    const float* conv_w = (const float*)d_in[6];
    const float* conv_b = (const float*)d_in[7];
    const float* bn_g   = (const float*)d_in[8];
    const float* bn_b   = (const float*)d_in[9];
    const float* out_w  = (const float*)d_in[10];
    const float* out_b  = (const float*)d_in[11];
    float* out = (float*)d_out;
    float* ws  = (float*)d_ws;

    qkv_kernel  <<<dim3(Ec / 16, (Bc * Sc) / 16, 3), 32,  0, stream>>>(x, Wq, Wk, Wv, ws);
    attn_kernel <<<dim3(LKc / 16, Bc * Hc, 4),      256, 0, stream>>>(ws, out);
    ln_kernel   <<<dim3(Bc * LCc),                  256, 0, stream>>>(ws, ln_g, ln_b);
    conv_kernel <<<dim3(Ec / 16, LCc / 16, Bc),     32,  0, stream>>>(conv_w, conv_b, ws);
    bnstat_kernel<<<dim3(Ec),                       256, 0, stream>>>(ws);
    pool_kernel <<<dim3(Sc, Bc),                    256, 0, stream>>>(bn_g, bn_b, ws, out);
    head_kernel <<<dim3(Bc),                        256, 0, stream>>>(out_w, out_b, out);
}